// NeuralMemory_19834158973335
// MI455X (gfx1250) — compile-verified
//
#include <hip/hip_runtime.h>
#include <math.h>

#define DIMD   256
#define HIDD   1024
#define CHUNKD 64
#define NTOK   2048
#define BATCHD 2
#define NCHUNK 32                   // NTOK / CHUNK
#define ROWS   (BATCHD * NTOK)      // 4096 tokens total
#define ZCH    (BATCHD * NCHUNK)    // 64 (b,chunk) pairs
#define EPSV   1e-6f

typedef __attribute__((ext_vector_type(2))) float v2f;
typedef __attribute__((ext_vector_type(8))) float v8f;

__device__ __forceinline__ float gelu_f(float x) {
  return 0.5f * x * (1.0f + erff(x * 0.70710678118654752f));
}
__device__ __forceinline__ float gelu_grad_f(float x) {
  float cdf = 0.5f * (1.0f + erff(x * 0.70710678118654752f));
  float pdf = 0.3989422804014327f * expf(-0.5f * x * x);
  return cdf + x * pdf;
}
__device__ __forceinline__ float sigmoid_f(float x) { return 1.0f / (1.0f + expf(-x)); }

__device__ __forceinline__ float wave_sum(float v) {
#pragma unroll
  for (int off = 16; off > 0; off >>= 1) v += __shfl_xor(v, off, 32);
  return v;
}

// ---------------------------------------------------------------------------
// f32 WMMA GEMM: C[M,N] = op(A)[M,K] @ op(B)[K,N]   (V_WMMA_F32_16X16X4_F32)
// TA/TB: operand stored transposed. AOP=1: gelu applied to A on load.
// EPI: 0 none | 1 gelu(C) | 2 C *= gelu'(Aux) (Aux has C's layout).
// Batched via blockIdx.z. Block = 256 threads = 8 waves (2x4 wave grid).
// Wave tile 32x32 (2x2 accumulators); block tile 64(M) x 128(N).
// Per k-step: 4 fragment loads feed 4 WMMAs (A/B fragments each reused twice).
// ---------------------------------------------------------------------------
template <bool TA, bool TB, int AOP, int EPI>
__global__ __launch_bounds__(256) void gemm_wmma(
    const float* __restrict__ Ab, const float* __restrict__ Bb,
    float* __restrict__ Cb, const float* __restrict__ Auxb,
    int M, int N, int K, int lda, int ldb, int ldc,
    long long sA, long long sB, long long sC, long long sAux) {
  const int z = blockIdx.z;
  const float* A = Ab + (size_t)z * (size_t)sA;
  const float* B = Bb + (size_t)z * (size_t)sB;
  float* C = Cb + (size_t)z * (size_t)sC;
  const float* Aux = (EPI == 2) ? (Auxb + (size_t)z * (size_t)sAux) : nullptr;

  const int wave = threadIdx.x >> 5;
  const int lane = threadIdx.x & 31;
  const int half = lane >> 4;  // 0: K pair {0,1}; 1: K pair {2,3}
  const int l16  = lane & 15;

  const int m0 = blockIdx.y * 64 + (wave >> 2) * 32;   // wave covers M rows [m0, m0+32)
  const int n0 = blockIdx.x * 128 + (wave & 3) * 32;   // wave covers N cols [n0, n0+32)
  if (m0 >= M || n0 >= N) return;  // wave-uniform: EXEC stays all-ones for WMMA

  v8f acc00 = {0.f,0.f,0.f,0.f,0.f,0.f,0.f,0.f};
  v8f acc01 = acc00, acc10 = acc00, acc11 = acc00;

  // per-lane base pointers; advanced by a constant stride each k-step
  const int am0 = m0 + l16, am1 = m0 + 16 + l16;
  const int bn0 = n0 + l16, bn1 = n0 + 16 + l16;
  const float* pA0 = TA ? (A + (size_t)(half * 2) * lda + am0)
                        : (A + (size_t)am0 * lda + half * 2);
  const float* pA1 = TA ? (A + (size_t)(half * 2) * lda + am1)
                        : (A + (size_t)am1 * lda + half * 2);
  const float* pB0 = TB ? (B + (size_t)bn0 * ldb + half * 2)
                        : (B + (size_t)(half * 2) * ldb + bn0);
  const float* pB1 = TB ? (B + (size_t)bn1 * ldb + half * 2)
                        : (B + (size_t)(half * 2) * ldb + bn1);
  const size_t stepA = TA ? (size_t)4 * lda : 4;   // advance per k-step (elements)
  const size_t stepB = TB ? 4 : (size_t)4 * ldb;
  const size_t dupA  = TA ? (size_t)lda : 1;       // offset of second element in pair
  const size_t dupB  = TB ? 1 : (size_t)ldb;

#pragma unroll 2
  for (int k0 = 0; k0 < K; k0 += 4) {
    v2f a0, a1, b0, b1;
    a0.x = pA0[0]; a0.y = pA0[dupA];
    a1.x = pA1[0]; a1.y = pA1[dupA];
    if (AOP == 1) {
      a0.x = gelu_f(a0.x); a0.y = gelu_f(a0.y);
      a1.x = gelu_f(a1.x); a1.y = gelu_f(a1.y);
    }
    b0.x = pB0[0]; b0.y = pB0[dupB];
    b1.x = pB1[0]; b1.y = pB1[dupB];
    acc00 = __builtin_amdgcn_wmma_f32_16x16x4_f32(false, a0, false, b0, (short)0, acc00, false, false);
    acc01 = __builtin_amdgcn_wmma_f32_16x16x4_f32(false, a0, false, b1, (short)0, acc01, false, false);
    acc10 = __builtin_amdgcn_wmma_f32_16x16x4_f32(false, a1, false, b0, (short)0, acc10, false, false);
    acc11 = __builtin_amdgcn_wmma_f32_16x16x4_f32(false, a1, false, b1, (short)0, acc11, false, false);
    pA0 += stepA; pA1 += stepA; pB0 += stepB; pB1 += stepB;
  }

#pragma unroll
  for (int rr = 0; rr < 8; ++rr) {
    // C VGPR r: lanes 0-15 -> M=r, lanes 16-31 -> M=r+8
    const int row0 = m0 + rr + half * 8;
    const int row1 = row0 + 16;
    const int col0 = n0 + l16;
    const int col1 = col0 + 16;
    float c00 = acc00[rr], c01 = acc01[rr], c10 = acc10[rr], c11 = acc11[rr];
    if (EPI == 1) {
      c00 = gelu_f(c00); c01 = gelu_f(c01); c10 = gelu_f(c10); c11 = gelu_f(c11);
    } else if (EPI == 2) {
      c00 *= gelu_grad_f(Aux[(size_t)row0 * ldc + col0]);
      c01 *= gelu_grad_f(Aux[(size_t)row0 * ldc + col1]);
      c10 *= gelu_grad_f(Aux[(size_t)row1 * ldc + col0]);
      c11 *= gelu_grad_f(Aux[(size_t)row1 * ldc + col1]);
    }
    C[(size_t)row0 * ldc + col0] = c00;
    C[(size_t)row0 * ldc + col1] = c01;
    C[(size_t)row1 * ldc + col0] = c10;
    C[(size_t)row1 * ldc + col1] = c11;
  }
}

// --------------------- rowwise rmsnorm of seq (both gains) ------------------
__global__ __launch_bounds__(256) void norm_kernel(
    const float* __restrict__ seq, const float* __restrict__ g_store,
    const float* __restrict__ g_ret, float* __restrict__ store,
    float* __restrict__ qin) {
  const int wave = threadIdx.x >> 5, lane = threadIdx.x & 31;
  const int row = blockIdx.x * 8 + wave;
  const size_t rb = (size_t)row * DIMD;
  float v[8], ss = 0.f;
#pragma unroll
  for (int j = 0; j < 8; ++j) { v[j] = seq[rb + j * 32 + lane]; ss += v[j] * v[j]; }
  ss = wave_sum(ss);
  const float s = rsqrtf(ss * (1.0f / DIMD) + EPSV);
#pragma unroll
  for (int j = 0; j < 8; ++j) {
    const int d = j * 32 + lane;
    const float nv = v[j] * s;
    store[rb + d] = nv * g_store[d];
    qin[rb + d]   = nv * g_ret[d];
  }
}

// ------------- per-chunk: lr per token, pooled -> momentum/decay gates ------
__global__ __launch_bounds__(256) void chunkstats_kernel(
    const float* __restrict__ store, const float* __restrict__ Wstep,
    const float* __restrict__ bstep, const float* __restrict__ Wmom,
    const float* __restrict__ bmom, const float* __restrict__ Wdec,
    const float* __restrict__ bdec, float* __restrict__ lr,
    float* __restrict__ gm, float* __restrict__ gd) {
  __shared__ float red[256];
  const int z = blockIdx.x, tid = threadIdx.x;
  const size_t base = (size_t)z * CHUNKD * DIMD;
  float pd = 0.f;
  for (int t = 0; t < CHUNKD; ++t) pd += store[base + (size_t)t * DIMD + tid];
  pd *= (1.0f / CHUNKD);

  red[tid] = pd * Wmom[tid];
  __syncthreads();
  for (int s2 = 128; s2 > 0; s2 >>= 1) {
    if (tid < s2) red[tid] += red[tid + s2];
    __syncthreads();
  }
  if (tid == 0) gm[z] = sigmoid_f(red[0] + bmom[0]);
  __syncthreads();
  red[tid] = pd * Wdec[tid];
  __syncthreads();
  for (int s2 = 128; s2 > 0; s2 >>= 1) {
    if (tid < s2) red[tid] += red[tid + s2];
    __syncthreads();
  }
  if (tid == 0) gd[z] = sigmoid_f(red[0] + bdec[0]);

  const int wave = tid >> 5, lane = tid & 31;
  for (int ti = 0; ti < 8; ++ti) {
    const int t = wave * 8 + ti;
    float dotv = 0.f;
#pragma unroll
    for (int j = 0; j < 8; ++j) {
      const int d = j * 32 + lane;
      dotv += store[base + (size_t)t * DIMD + d] * Wstep[d];
    }
    dotv = wave_sum(dotv);
    if (lane == 0) lr[z * CHUNKD + t] = sigmoid_f(dotv + bstep[0]);
  }
}

// ---- loss backward: out = rmsnorm(y)*rn + k ; dout -> dy (rmsnorm bwd), grad_rn
__global__ __launch_bounds__(256) void dy_kernel(
    const float* __restrict__ y, const float* __restrict__ kbuf,
    const float* __restrict__ vbuf, const float* __restrict__ lr,
    const float* __restrict__ rn_g, float* __restrict__ dy,
    float* __restrict__ grn) {
  __shared__ float accs[DIMD];
  const int z = blockIdx.x, tid = threadIdx.x;
  accs[tid] = 0.f;
  __syncthreads();
  const int wave = tid >> 5, lane = tid & 31;
  for (int ti = 0; ti < 8; ++ti) {
    const int t = wave * 8 + ti;
    const size_t rb = ((size_t)z * CHUNKD + t) * DIMD;
    float yv[8], kv[8], vv[8], ss = 0.f;
#pragma unroll
    for (int j = 0; j < 8; ++j) {
      const int d = j * 32 + lane;
      yv[j] = y[rb + d]; kv[j] = kbuf[rb + d]; vv[j] = vbuf[rb + d];
      ss += yv[j] * yv[j];
    }
    ss = wave_sum(ss);
    const float s = rsqrtf(ss * (1.0f / DIMD) + EPSV);
    const float co = 2.0f * lr[z * CHUNKD + t] * (1.0f / DIMD);
    float dr[8], dotry = 0.f;
#pragma unroll
    for (int j = 0; j < 8; ++j) {
      const int d = j * 32 + lane;
      const float r = yv[j] * s;
      const float o = r * rn_g[d] + kv[j];
      const float dout = co * (o - vv[j]);
      atomicAdd(&accs[d], dout * r);  // grad_rn accumulation (ds_add_f32)
      dr[j] = dout * rn_g[d];
      dotry += dr[j] * yv[j];
    }
    dotry = wave_sum(dotry);
    const float coef = dotry * s * s * (1.0f / DIMD);
#pragma unroll
    for (int j = 0; j < 8; ++j) {
      const int d = j * 32 + lane;
      dy[rb + d] = s * (dr[j] - yv[j] * coef);
    }
  }
  __syncthreads();
  grn[(size_t)z * DIMD + tid] = accs[tid];
}

// ------- momentum+decay scan over chunks; emits shifted "used" weights ------
__global__ __launch_bounds__(256) void scan_kernel(
    const float* __restrict__ grad, const float* __restrict__ w0,
    const float* __restrict__ gm, const float* __restrict__ gd,
    float* __restrict__ used, long long P) {
  const long long idx = (long long)blockIdx.x * blockDim.x + threadIdx.x;
  if (idx >= (long long)BATCHD * P) return;
  const long long b = idx / P, i = idx % P;
  float W = w0[i], m = 0.f;
  for (int t = 0; t < NCHUNK; ++t) {
    const long long zz = b * NCHUNK + t;
    used[(size_t)zz * P + i] = W;               // weights after chunks < t
    m = gm[zz] * m - grad[(size_t)zz * P + i];  // surprise = -grad
    W = (1.0f - gd[zz]) * W + m;
  }
}

// ------------- retrieval epilogue: rmsnorm(y2)*rnu + q -> out ---------------
__global__ __launch_bounds__(256) void out_kernel(
    const float* __restrict__ y2, const float* __restrict__ q,
    const float* __restrict__ rnu, float* __restrict__ out) {
  const int z = blockIdx.x, tid = threadIdx.x;
  const int wave = tid >> 5, lane = tid & 31;
  for (int ti = 0; ti < 8; ++ti) {
    const int t = wave * 8 + ti;
    const size_t rb = ((size_t)z * CHUNKD + t) * DIMD;
    float yv[8], ss = 0.f;
#pragma unroll
    for (int j = 0; j < 8; ++j) { yv[j] = y2[rb + j * 32 + lane]; ss += yv[j] * yv[j]; }
    ss = wave_sum(ss);
    const float s = rsqrtf(ss * (1.0f / DIMD) + EPSV);
#pragma unroll
    for (int j = 0; j < 8; ++j) {
      const int d = j * 32 + lane;
      out[rb + d] = yv[j] * s * rnu[(size_t)z * DIMD + d] + q[rb + d];
    }
  }
}

// ---------------------------------------------------------------------------
extern "C" void kernel_launch(void* const* d_in, const int* in_sizes, int n_in,
                              void* d_out, int out_size, void* d_ws, size_t ws_size,
                              hipStream_t stream) {
  (void)in_sizes; (void)n_in; (void)out_size; (void)ws_size;
  const float* seq     = (const float*)d_in[0];
  const float* g_store = (const float*)d_in[1];
  const float* g_ret   = (const float*)d_in[2];
  const float* Wq      = (const float*)d_in[3];
  const float* Wk      = (const float*)d_in[4];
  const float* Wv      = (const float*)d_in[5];
  const float* Wstep   = (const float*)d_in[6];
  const float* bstep   = (const float*)d_in[7];
  const float* Wmom    = (const float*)d_in[8];
  const float* bmom    = (const float*)d_in[9];
  const float* Wdec    = (const float*)d_in[10];
  const float* bdec    = (const float*)d_in[11];
  const float* w1      = (const float*)d_in[12];
  const float* w2      = (const float*)d_in[13];
  const float* rn_g    = (const float*)d_in[14];
  float* out = (float*)d_out;

  // workspace layout (float elements)
  constexpr size_t SZ_RD = (size_t)ROWS * DIMD;   // 1,048,576
  constexpr size_t SZ_RH = (size_t)ROWS * HIDD;   // 4,194,304
  constexpr size_t SZ_P  = (size_t)DIMD * HIDD;   // 262,144
  constexpr size_t SZ_GW = (size_t)ZCH * SZ_P;    // 16,777,216
  float* w = (float*)d_ws;
  size_t off = 0;
  float* ws_store = w + off; off += SZ_RD;
  float* ws_qin   = w + off; off += SZ_RD;
  float* ws_k     = w + off; off += SZ_RD;
  float* ws_v     = w + off; off += SZ_RD;
  float* ws_q     = w + off; off += SZ_RD;
  float* ws_lr    = w + off; off += (size_t)ROWS;
  float* ws_gm    = w + off; off += (size_t)ZCH;
  float* ws_gd    = w + off; off += (size_t)ZCH;
  float* ws_h     = w + off; off += SZ_RH;
  float* ws_y     = w + off; off += SZ_RD;
  float* ws_dy    = w + off; off += SZ_RD;
  float* ws_dh    = w + off; off += SZ_RH;
  float* ws_gw1   = w + off; off += SZ_GW;
  float* ws_gw2   = w + off; off += SZ_GW;
  float* ws_grn   = w + off; off += (size_t)ZCH * DIMD;
  float* ws_w1u   = w + off; off += SZ_GW;
  float* ws_w2u   = w + off; off += SZ_GW;
  float* ws_rnu   = w + off; off += (size_t)ZCH * DIMD;
  float* ws_hg2   = w + off; off += SZ_RH;
  float* ws_y2    = w + off; off += SZ_RD;

  const dim3 blk(256);
  // 1) rmsnorm (store + retrieve gains)
  norm_kernel<<<ROWS / 8, blk, 0, stream>>>(seq, g_store, g_ret, ws_store, ws_qin);

  // 2) projections k, v, q  ([4096,256]@[256,256])
  {
    dim3 g(DIMD / 128, ROWS / 64, 1);
    gemm_wmma<false, false, 0, 0><<<g, blk, 0, stream>>>(ws_store, Wk, ws_k, nullptr,
        ROWS, DIMD, DIMD, DIMD, DIMD, DIMD, 0, 0, 0, 0);
    gemm_wmma<false, false, 0, 0><<<g, blk, 0, stream>>>(ws_store, Wv, ws_v, nullptr,
        ROWS, DIMD, DIMD, DIMD, DIMD, DIMD, 0, 0, 0, 0);
    gemm_wmma<false, false, 0, 0><<<g, blk, 0, stream>>>(ws_qin, Wq, ws_q, nullptr,
        ROWS, DIMD, DIMD, DIMD, DIMD, DIMD, 0, 0, 0, 0);
  }

  // 3) per-token lr + per-chunk gates
  chunkstats_kernel<<<ZCH, blk, 0, stream>>>(ws_store, Wstep, bstep, Wmom, bmom,
                                             Wdec, bdec, ws_lr, ws_gm, ws_gd);

  // 4) inner forward: h = k@w1 (pre-activation kept for backward)
  {
    dim3 g(HIDD / 128, ROWS / 64, 1);
    gemm_wmma<false, false, 0, 0><<<g, blk, 0, stream>>>(ws_k, w1, ws_h, nullptr,
        ROWS, HIDD, DIMD, DIMD, HIDD, HIDD, 0, 0, 0, 0);
  }
  // 5) y = gelu(h)@w2
  {
    dim3 g(DIMD / 128, ROWS / 64, 1);
    gemm_wmma<false, false, 1, 0><<<g, blk, 0, stream>>>(ws_h, w2, ws_y, nullptr,
        ROWS, DIMD, HIDD, HIDD, DIMD, DIMD, 0, 0, 0, 0);
  }
  // 6) loss + rmsnorm backward -> dy, grad_rn
  dy_kernel<<<ZCH, blk, 0, stream>>>(ws_y, ws_k, ws_v, ws_lr, rn_g, ws_dy, ws_grn);

  // 7) dh = (dy @ w2^T) * gelu'(h)
  {
    dim3 g(HIDD / 128, ROWS / 64, 1);
    gemm_wmma<false, true, 0, 2><<<g, blk, 0, stream>>>(ws_dy, w2, ws_dh, ws_h,
        ROWS, HIDD, DIMD, DIMD, DIMD, HIDD, 0, 0, 0, 0);
  }
  // 8) grad w2 per chunk: gelu(h_c)^T @ dy_c   [1024,64]@[64,256], batch=64
  {
    dim3 g(DIMD / 128, HIDD / 64, ZCH);
    gemm_wmma<true, false, 1, 0><<<g, blk, 0, stream>>>(ws_h, ws_dy, ws_gw2, nullptr,
        HIDD, DIMD, CHUNKD, HIDD, DIMD, DIMD,
        (long long)CHUNKD * HIDD, (long long)CHUNKD * DIMD, (long long)HIDD * DIMD, 0);
  }
  // 9) grad w1 per chunk: k_c^T @ dh_c   [256,64]@[64,1024], batch=64
  {
    dim3 g(HIDD / 128, DIMD / 64, ZCH);
    gemm_wmma<true, false, 0, 0><<<g, blk, 0, stream>>>(ws_k, ws_dh, ws_gw1, nullptr,
        DIMD, HIDD, CHUNKD, DIMD, HIDD, HIDD,
        (long long)CHUNKD * DIMD, (long long)CHUNKD * HIDD, (long long)DIMD * HIDD, 0);
  }

  // 10) momentum/decay scans -> shifted used weights
  scan_kernel<<<(unsigned)((BATCHD * SZ_P + 255) / 256), blk, 0, stream>>>(
      ws_gw1, w1, ws_gm, ws_gd, ws_w1u, (long long)SZ_P);
  scan_kernel<<<(unsigned)((BATCHD * SZ_P + 255) / 256), blk, 0, stream>>>(
      ws_gw2, w2, ws_gm, ws_gd, ws_w2u, (long long)SZ_P);
  scan_kernel<<<(unsigned)((BATCHD * DIMD + 255) / 256), blk, 0, stream>>>(
      ws_grn, rn_g, ws_gm, ws_gd, ws_rnu, (long long)DIMD);

  // 11) retrieval: hg2 = gelu(q_c @ w1u_c)   [64,256]@[256,1024], batch=64
  {
    dim3 g(HIDD / 128, CHUNKD / 64, ZCH);
    gemm_wmma<false, false, 0, 1><<<g, blk, 0, stream>>>(ws_q, ws_w1u, ws_hg2, nullptr,
        CHUNKD, HIDD, DIMD, DIMD, HIDD, HIDD,
        (long long)CHUNKD * DIMD, (long long)SZ_P, (long long)CHUNKD * HIDD, 0);
  }
  // 12) y2 = hg2 @ w2u_c   [64,1024]@[1024,256], batch=64
  {
    dim3 g(DIMD / 128, CHUNKD / 64, ZCH);
    gemm_wmma<false, false, 0, 0><<<g, blk, 0, stream>>>(ws_hg2, ws_w2u, ws_y2, nullptr,
        CHUNKD, DIMD, HIDD, HIDD, DIMD, DIMD,
        (long long)CHUNKD * HIDD, (long long)SZ_P, (long long)CHUNKD * DIMD, 0);
  }
  // 13) out = rmsnorm(y2)*rnu + q
  out_kernel<<<ZCH, blk, 0, stream>>>(ws_y2, ws_q, ws_rnu, out);
}